// Flow_Attention_41274635715038
// MI455X (gfx1250) — compile-verified
//
#include <hip/hip_runtime.h>
#include <math.h>

// Problem shape (fixed by the reference): N=4, L=16384, H=8, D=16
#define NB      4
#define SEQ     16384
#define NHEAD   8
#define DIM     16
#define HEADS   (NB * NHEAD)     // 32 (n,h) pairs
#define LSTRIDE (NHEAD * DIM)    // 128 floats between consecutive l for fixed (n,h)

#define EPSF    1e-6f
#define LN_EPSF 1e-5f

// Compile-time grid/chunk configuration (kernel_launch must match)
#define GRID_SUM 64
#define CH_SUM   (SEQ / GRID_SUM)      // 256 rows per block, 4 iters per lane-group
#define GRID_KV  16
#define CH_KV    (SEQ / GRID_KV)       // 1024 rows per block
#define PW_KV    (CH_KV / 8)           // 128 rows per wave
#define GRID_OUT 32
#define CH_OUT   (SEQ / GRID_OUT)      // 512 rows per block
#define PW_OUT   (CH_OUT / 8)          // 64 rows per wave (4 tiles of 16)

// Workspace layout (floats)
#define WS_KSUM 0                          // [HEADS][DIM] sum_s sigmoid(k)
#define WS_QSUM (WS_KSUM + HEADS * DIM)    // [HEADS][DIM] sum_l sigmoid(q)
#define WS_KN   (WS_QSUM + HEADS * DIM)    // [HEADS][DIM] sum_s k*ncol
#define WS_QN   (WS_KN + HEADS * DIM)      // [HEADS][DIM] sum_l q*nrow
#define WS_Z    (WS_QN + HEADS * DIM)      // [HEADS] softmax denominator (padded to 64)
#define WS_KV   (WS_Z + 64)                // [HEADS][DIM][DIM] kv matrices
#define WS_TOTAL (WS_KV + HEADS * DIM * DIM)

typedef float v2f __attribute__((ext_vector_type(2)));
typedef float v8f __attribute__((ext_vector_type(8)));

__device__ __forceinline__ float sigmoidf_(float x) { return 1.0f / (1.0f + __expf(-x)); }

__device__ __forceinline__ int headBase(int nh) {
    int n = nh >> 3;          // / NHEAD
    int h = nh & 7;           // % NHEAD
    return n * (SEQ * LSTRIDE) + h * DIM;
}

__device__ __forceinline__ float4 sig4(float4 v) {
    float4 r;
    r.x = sigmoidf_(v.x); r.y = sigmoidf_(v.y);
    r.z = sigmoidf_(v.z); r.w = sigmoidf_(v.w);
    return r;
}

// ---------------------------------------------------------------------------
// Phase 1: ksum[d] = sum_s sigmoid(K), qsum[d] = sum_l sigmoid(Q)
// b128 loads: lane = (row-group, d-quad); 8 rows / 512B per wave per load.
// ---------------------------------------------------------------------------
__global__ void k_sums(const float* __restrict__ Q, const float* __restrict__ K,
                       float* __restrict__ ws) {
    int nh = blockIdx.y;
    int base = headBase(nh);
    int dq = threadIdx.x & 3;           // d quad: d in [4*dq, 4*dq+4)
    int rg = threadIdx.x >> 2;          // 0..63 row group
    int l0 = blockIdx.x * CH_SUM;

    float ks0 = 0.f, ks1 = 0.f, ks2 = 0.f, ks3 = 0.f;
    float qs0 = 0.f, qs1 = 0.f, qs2 = 0.f, qs3 = 0.f;
    for (int l = l0 + rg; l < l0 + CH_SUM; l += 64) {
        int idx = base + l * LSTRIDE + 4 * dq;
        float4 kf = sig4(*(const float4*)(K + idx));
        float4 qf = sig4(*(const float4*)(Q + idx));
        ks0 += kf.x; ks1 += kf.y; ks2 += kf.z; ks3 += kf.w;
        qs0 += qf.x; qs1 += qf.y; qs2 += qf.z; qs3 += qf.w;
    }
    __shared__ float sks[DIM], sqs[DIM];
    if (threadIdx.x < DIM) { sks[threadIdx.x] = 0.f; sqs[threadIdx.x] = 0.f; }
    __syncthreads();
    atomicAdd(&sks[4 * dq + 0], ks0); atomicAdd(&sks[4 * dq + 1], ks1);
    atomicAdd(&sks[4 * dq + 2], ks2); atomicAdd(&sks[4 * dq + 3], ks3);
    atomicAdd(&sqs[4 * dq + 0], qs0); atomicAdd(&sqs[4 * dq + 1], qs1);
    atomicAdd(&sqs[4 * dq + 2], qs2); atomicAdd(&sqs[4 * dq + 3], qs3);
    __syncthreads();
    if (threadIdx.x < DIM) {
        atomicAdd(&ws[WS_KSUM + nh * DIM + threadIdx.x], sks[threadIdx.x]);
        atomicAdd(&ws[WS_QSUM + nh * DIM + threadIdx.x], sqs[threadIdx.x]);
    }
}

// ---------------------------------------------------------------------------
// Phase 2: kn[d] = sum_s k_sig / dot(k_sig+eps, qsum+eps)
//          qn[d] = sum_l q_sig / dot(q_sig+eps, ksum+eps)
// Row dots: per-lane 4-term FMA + 2-step shfl_xor butterfly over 4 lanes.
// ---------------------------------------------------------------------------
__global__ void k_refine_sums(const float* __restrict__ Q, const float* __restrict__ K,
                              float* __restrict__ ws) {
    int nh = blockIdx.y;
    int base = headBase(nh);
    int dq = threadIdx.x & 3;
    int rg = threadIdx.x >> 2;
    int l0 = blockIdx.x * CH_SUM;

    float4 qsum4 = *(const float4*)(ws + WS_QSUM + nh * DIM + 4 * dq);
    float4 ksum4 = *(const float4*)(ws + WS_KSUM + nh * DIM + 4 * dq);
    qsum4.x += EPSF; qsum4.y += EPSF; qsum4.z += EPSF; qsum4.w += EPSF;
    ksum4.x += EPSF; ksum4.y += EPSF; ksum4.z += EPSF; ksum4.w += EPSF;

    float kn0 = 0.f, kn1 = 0.f, kn2 = 0.f, kn3 = 0.f;
    float qn0 = 0.f, qn1 = 0.f, qn2 = 0.f, qn3 = 0.f;
    for (int l = l0 + rg; l < l0 + CH_SUM; l += 64) {
        int idx = base + l * LSTRIDE + 4 * dq;
        float4 kf = sig4(*(const float4*)(K + idx));
        float4 qf = sig4(*(const float4*)(Q + idx));
        float dk = (kf.x + EPSF) * qsum4.x + (kf.y + EPSF) * qsum4.y +
                   (kf.z + EPSF) * qsum4.z + (kf.w + EPSF) * qsum4.w;
        float dqt = (qf.x + EPSF) * ksum4.x + (qf.y + EPSF) * ksum4.y +
                    (qf.z + EPSF) * ksum4.z + (qf.w + EPSF) * ksum4.w;
        dk  += __shfl_xor(dk, 1, 4);  dk  += __shfl_xor(dk, 2, 4);
        dqt += __shfl_xor(dqt, 1, 4); dqt += __shfl_xor(dqt, 2, 4);
        float rk = 1.0f / dk, rq = 1.0f / dqt;
        kn0 += kf.x * rk; kn1 += kf.y * rk; kn2 += kf.z * rk; kn3 += kf.w * rk;
        qn0 += qf.x * rq; qn1 += qf.y * rq; qn2 += qf.z * rq; qn3 += qf.w * rq;
    }
    __shared__ float skn[DIM], sqn[DIM];
    if (threadIdx.x < DIM) { skn[threadIdx.x] = 0.f; sqn[threadIdx.x] = 0.f; }
    __syncthreads();
    atomicAdd(&skn[4 * dq + 0], kn0); atomicAdd(&skn[4 * dq + 1], kn1);
    atomicAdd(&skn[4 * dq + 2], kn2); atomicAdd(&skn[4 * dq + 3], kn3);
    atomicAdd(&sqn[4 * dq + 0], qn0); atomicAdd(&sqn[4 * dq + 1], qn1);
    atomicAdd(&sqn[4 * dq + 2], qn2); atomicAdd(&sqn[4 * dq + 3], qn3);
    __syncthreads();
    if (threadIdx.x < DIM) {
        atomicAdd(&ws[WS_KN + nh * DIM + threadIdx.x], skn[threadIdx.x]);
        atomicAdd(&ws[WS_QN + nh * DIM + threadIdx.x], sqn[threadIdx.x]);
    }
}

// ---------------------------------------------------------------------------
// Phase 3: Z = sum_s exp(dot(k_sig+eps, qn+eps))   (softmax denominator)
// ---------------------------------------------------------------------------
__global__ void k_softmax_den(const float* __restrict__ K, float* __restrict__ ws) {
    int nh = blockIdx.y;
    int base = headBase(nh);
    int dq = threadIdx.x & 3;
    int rg = threadIdx.x >> 2;
    int l0 = blockIdx.x * CH_SUM;

    float4 qn4 = *(const float4*)(ws + WS_QN + nh * DIM + 4 * dq);
    qn4.x += EPSF; qn4.y += EPSF; qn4.z += EPSF; qn4.w += EPSF;

    float z = 0.f;
    for (int l = l0 + rg; l < l0 + CH_SUM; l += 64) {
        float4 kf = sig4(*(const float4*)(K + base + l * LSTRIDE + 4 * dq));
        float t = (kf.x + EPSF) * qn4.x + (kf.y + EPSF) * qn4.y +
                  (kf.z + EPSF) * qn4.z + (kf.w + EPSF) * qn4.w;
        t += __shfl_xor(t, 1, 4);
        t += __shfl_xor(t, 2, 4);
        if (dq == 0) z += __expf(t);
    }
    __shared__ float sz;
    if (threadIdx.x == 0) sz = 0.f;
    __syncthreads();
    if (dq == 0) atomicAdd(&sz, z);
    __syncthreads();
    if (threadIdx.x == 0) atomicAdd(&ws[WS_Z + nh], sz);
}

// ---------------------------------------------------------------------------
// Phase 4: kv[d][e] = sum_s k_sig[s][d] * (v[s][e] * cr[s]) via
// v_wmma_f32_16x16x4_f32: A = K^T tile (16d x 4s), B = Vc tile (4s x 16e).
// A layout (ISA 7.12.2): lane(M=lane&15, half=lane>>4), VGPR j -> K = 2*half+j.
// Each wave accumulates a private 16x16 C over its sequence chunk.
// ---------------------------------------------------------------------------
__global__ void k_kv_wmma(const float* __restrict__ K, const float* __restrict__ V,
                          float* __restrict__ ws) {
    int nh = blockIdx.y;
    int base = headBase(nh);
    int lane = threadIdx.x & 31;
    int wave = threadIdx.x >> 5;     // 0..7
    int half = lane >> 4;
    int m = lane & 15;

    float qn_d = ws[WS_QN + nh * DIM + m] + EPSF;
    float scale = (float)SEQ / ws[WS_Z + nh];   // softmax * S

    int s0 = blockIdx.x * CH_KV + wave * PW_KV;

    v8f acc = {};
    for (int s = s0; s < s0 + PW_KV; s += 4) {
        int p0 = s + 2 * half;                 // this half's two sequence positions
        int i0 = base + p0 * LSTRIDE + m;
        float k0 = sigmoidf_(K[i0]);
        float k1 = sigmoidf_(K[i0 + LSTRIDE]);
        // col-refine for p0/p0+1: dot over d (16 lanes of this half)
        float t0 = (k0 + EPSF) * qn_d;
        float t1 = (k1 + EPSF) * qn_d;
        #pragma unroll
        for (int x = 1; x < 16; x <<= 1) {
            t0 += __shfl_xor(t0, x, 16);
            t1 += __shfl_xor(t1, x, 16);
        }
        float cr0 = __expf(t0) * scale;
        float cr1 = __expf(t1) * scale;

        v2f a; a.x = k0;           a.y = k1;                    // A: K^T (d x s)
        v2f b; b.x = V[i0] * cr0;  b.y = V[i0 + LSTRIDE] * cr1; // B: Vc (s x e)
        acc = __builtin_amdgcn_wmma_f32_16x16x4_f32(
            false, a, false, b, (short)0, acc, false, false);
    }
    float* kvg = &ws[WS_KV + nh * (DIM * DIM)];
    #pragma unroll
    for (int r = 0; r < 8; ++r)
        atomicAdd(&kvg[(r + 8 * half) * DIM + m], acc[r]);
}

// ---------------------------------------------------------------------------
// Phase 5: per 16-row tile: X = Q_sig(16x16) @ kv(16x16) via 4 chained
// v_wmma_f32_16x16x4_f32 (kv hoisted into registers), row-scale by
// nrow * sigmoid(nrow_refine), add residual V, LayerNorm over D=16, store.
// ---------------------------------------------------------------------------
__global__ void k_out_wmma(const float* __restrict__ Q, const float* __restrict__ V,
                           const float* __restrict__ gamma, const float* __restrict__ beta,
                           const float* __restrict__ ws, float* __restrict__ out) {
    int nh = blockIdx.y;
    int base = headBase(nh);
    int lane = threadIdx.x & 31;
    int wave = threadIdx.x >> 5;
    int half = lane >> 4;
    int m = lane & 15;

    // Loop-invariant per-lane data: this lane's A-side d indices are
    // d = 4c + 2*half + j  (c chunk, j in {0,1}).
    float ksA[8], knA[8];
    v2f bv[4];
    #pragma unroll
    for (int c = 0; c < 4; ++c) {
        #pragma unroll
        for (int j = 0; j < 2; ++j) {
            int dd = 4 * c + 2 * half + j;
            ksA[2 * c + j] = ws[WS_KSUM + nh * DIM + dd] + EPSF;
            knA[2 * c + j] = ws[WS_KN + nh * DIM + dd] + EPSF;
            bv[c][j] = ws[WS_KV + nh * (DIM * DIM) + dd * DIM + m];   // B: kv(d x e)
        }
    }
    float gam = gamma[m], bet = beta[m];

    int l0w = blockIdx.x * CH_OUT + wave * PW_OUT;

    for (int lt = l0w; lt < l0w + PW_OUT; lt += 16) {
        int row = lt + m;   // this lane's A row (lanes m and m+16 share row, split d)
        v2f a[4];
        float dr = 0.f, drr = 0.f;
        #pragma unroll
        for (int c = 0; c < 4; ++c) {
            // two consecutive floats -> one b64 load
            v2f qv = *(const v2f*)(Q + base + row * LSTRIDE + (4 * c + 2 * half));
            float q0 = sigmoidf_(qv.x);
            float q1 = sigmoidf_(qv.y);
            a[c][0] = q0;
            a[c][1] = q1;
            dr  += (q0 + EPSF) * ksA[2 * c] + (q1 + EPSF) * ksA[2 * c + 1];
            drr += (q0 + EPSF) * knA[2 * c] + (q1 + EPSF) * knA[2 * c + 1];
        }
        // combine the two half-lane partials (each half holds 8 of 16 d's)
        dr  += __shfl_xor(dr, 16, 32);
        drr += __shfl_xor(drr, 16, 32);
        float rowscale = (1.0f / dr) * sigmoidf_(drr);   // L/S == 1

        v8f acc = {};
        #pragma unroll
        for (int c = 0; c < 4; ++c)
            acc = __builtin_amdgcn_wmma_f32_16x16x4_f32(
                false, a[c], false, bv[c], (short)0, acc, false, false);

        #pragma unroll
        for (int r = 0; r < 8; ++r) {
            int rrow = lt + r + 8 * half;                 // C row for this VGPR
            float rs = __shfl(rowscale, r + 8 * half, 32);
            int oidx = base + rrow * LSTRIDE + m;         // column e = lane&15
            float x = acc[r] * rs + V[oidx];              // residual add
            // LayerNorm over e: 16 columns live in the 16 lanes of this half
            float s = x;
            #pragma unroll
            for (int t = 1; t < 16; t <<= 1) s += __shfl_xor(s, t, 16);
            float mean = s * (1.0f / 16.0f);
            float diff = x - mean;
            float v2 = diff * diff;
            #pragma unroll
            for (int t = 1; t < 16; t <<= 1) v2 += __shfl_xor(v2, t, 16);
            float var = v2 * (1.0f / 16.0f);
            out[oidx] = diff * rsqrtf(var + LN_EPSF) * gam + bet;
        }
    }
}

// ---------------------------------------------------------------------------
extern "C" void kernel_launch(void* const* d_in, const int* in_sizes, int n_in,
                              void* d_out, int out_size, void* d_ws, size_t ws_size,
                              hipStream_t stream) {
    (void)in_sizes; (void)n_in; (void)out_size; (void)ws_size;
    const float* Q = (const float*)d_in[0];
    const float* K = (const float*)d_in[1];
    const float* V = (const float*)d_in[2];
    const float* g = (const float*)d_in[3];
    const float* b = (const float*)d_in[4];
    float* ws = (float*)d_ws;
    float* out = (float*)d_out;

    hipMemsetAsync(d_ws, 0, WS_TOTAL * sizeof(float), stream);

    dim3 blk(256);
    k_sums<<<dim3(GRID_SUM, HEADS), blk, 0, stream>>>(Q, K, ws);
    k_refine_sums<<<dim3(GRID_SUM, HEADS), blk, 0, stream>>>(Q, K, ws);
    k_softmax_den<<<dim3(GRID_SUM, HEADS), blk, 0, stream>>>(K, ws);
    k_kv_wmma<<<dim3(GRID_KV, HEADS), blk, 0, stream>>>(K, V, ws);
    k_out_wmma<<<dim3(GRID_OUT, HEADS), blk, 0, stream>>>(Q, V, g, b, ws, out);
}